// SplineDisp_3066606649943
// MI455X (gfx1250) — compile-verified
//
#include <hip/hip_runtime.h>
#include <math.h>

#define BLOCK 256
#define B0 2
#define B1 8
#define B2 16
#define LDS_CAP 6144   // floats = 24 KB per workgroup

#define GAS __attribute__((address_space(1)))
#define LAS __attribute__((address_space(3)))

__device__ __forceinline__ int wrapi(int v, int n) {
    int r = v % n;
    return (r < 0) ? r + n : r;
}

__device__ __forceinline__ void bw4(float t, float* w) {
    const float s = 1.0f / 6.0f;
    float t2 = t * t, t3 = t2 * t;
    w[0] = (1.0f - 3.0f * t + 3.0f * t2 - t3) * s;
    w[1] = (4.0f - 6.0f * t2 + 3.0f * t3) * s;
    w[2] = (1.0f + 3.0f * t + 3.0f * t2 - 3.0f * t3) * s;
    w[3] = t3 * s;
}

// out12 = rows 0..2 of inv(P) @ Q, for 4x4 row-major affine matrices (last row 0,0,0,1)
__device__ void affine_inv_mul(const float* P, const float* Q, float* out12) {
    float a = P[0], b = P[1], c = P[2];
    float d = P[4], e = P[5], f = P[6];
    float g = P[8], h = P[9], i = P[10];
    float A =  (e * i - f * h);
    float B = -(d * i - f * g);
    float C =  (d * h - e * g);
    float det = a * A + b * B + c * C;
    float id = 1.0f / det;
    float Ri[9];
    Ri[0] = A * id;              Ri[1] = -(b * i - c * h) * id;  Ri[2] =  (b * f - c * e) * id;
    Ri[3] = B * id;              Ri[4] =  (a * i - c * g) * id;  Ri[5] = -(a * f - c * d) * id;
    Ri[6] = C * id;              Ri[7] = -(a * h - b * g) * id;  Ri[8] =  (a * e - b * d) * id;
    float ti[3];
    for (int r = 0; r < 3; r++)
        ti[r] = -(Ri[r * 3 + 0] * P[3] + Ri[r * 3 + 1] * P[7] + Ri[r * 3 + 2] * P[11]);
    for (int r = 0; r < 3; r++) {
        for (int cq = 0; cq < 4; cq++) {
            float s = Ri[r * 3 + 0] * Q[0 * 4 + cq]
                    + Ri[r * 3 + 1] * Q[1 * 4 + cq]
                    + Ri[r * 3 + 2] * Q[2 * 4 + cq];
            if (cq == 3) s += ti[r];
            out12[r * 4 + cq] = s;
        }
    }
}

__global__ void setup_mats_kernel(const float* __restrict__ A, const float* __restrict__ F,
                                  const float* __restrict__ M, float* __restrict__ ws) {
    if (threadIdx.x == 0 && blockIdx.x == 0) {
        float fa[12], ma[12];
        affine_inv_mul(A, F, fa);   // fixed voxels -> field voxels
        affine_inv_mul(M, A, ma);   // field voxels -> moving voxels
        for (int k = 0; k < 12; k++) { ws[k] = fa[k]; ws[12 + k] = ma[k]; }
    }
}

__global__ __launch_bounds__(BLOCK) void spline_disp_kernel(
    const float* __restrict__ coeff, const float* __restrict__ mats,
    const int* __restrict__ pfd0, const int* __restrict__ pfd1, const int* __restrict__ pfd2,
    int X, int Y, int Z, float* __restrict__ out)
{
    __shared__ float tile[LDS_CAP];

    const int fd0 = *pfd0, fd1 = *pfd1, fd2 = *pfd2;

    float fa[12], ma[12];
#pragma unroll
    for (int k = 0; k < 12; k++) { fa[k] = mats[k]; ma[k] = mats[12 + k]; }

    const int nt0 = (fd0 + B0 - 1) / B0;
    const int nt1 = (fd1 + B1 - 1) / B1;
    const int nt2 = (fd2 + B2 - 1) / B2;
    const long long NT = (long long)nt0 * nt1 * nt2;

    const int tid = threadIdx.x;
    const int vi2 = tid & (B2 - 1);
    const int vi1 = (tid >> 4) & (B1 - 1);
    const int vi0 = tid >> 7;

    for (long long t = blockIdx.x; t < NT; t += gridDim.x) {
        int tt = (int)t;
        const int t2 = tt % nt2; tt /= nt2;
        const int t1 = tt % nt1;
        const int t0 = tt / nt1;
        const int o0 = t0 * B0, o1 = t1 * B1, o2 = t2 * B2;

        // Field-space bounding box of this output tile (affine => extrema at corners).
        float ming[3], maxg[3];
        const float lo[3] = { (float)o0, (float)o1, (float)o2 };
        const float hi[3] = { (float)(o0 + B0 - 1), (float)(o1 + B1 - 1), (float)(o2 + B2 - 1) };
#pragma unroll
        for (int k = 0; k < 3; k++) {
            float mn = fa[k * 4 + 3], mx = fa[k * 4 + 3];
#pragma unroll
            for (int j = 0; j < 3; j++) {
                float p = fa[k * 4 + j] * lo[j];
                float q = fa[k * 4 + j] * hi[j];
                mn += fminf(p, q);
                mx += fmaxf(p, q);
            }
            ming[k] = mn; maxg[k] = mx;
        }
        // Taps span [floor(g)-1, floor(g)+2]; +/-1 extra margin guards FP rounding.
        const int bx0 = (int)floorf(ming[0]) - 2;
        const int by0 = (int)floorf(ming[1]) - 2;
        const int bz0 = (int)floorf(ming[2]) - 2;
        const int Tx = (int)floorf(maxg[0]) - (int)floorf(ming[0]) + 6;
        const int Ty = (int)floorf(maxg[1]) - (int)floorf(ming[1]) + 6;
        const int Tz = (int)floorf(maxg[2]) - (int)floorf(ming[2]) + 6;
        const int nE = Tx * Ty * Tz * 3;
        const bool useLds = (nE <= LDS_CAP);   // uniform across block

        __syncthreads();   // previous iteration's LDS readers are done

        if (useLds) {
            // Async-stage wrapped coeff tile into LDS (per-lane addressed => wrap is free).
            for (int e = tid; e < nE; e += BLOCK) {
                int q = e / 3;
                const int c  = e - q * 3;
                const int dz = q % Tz; q /= Tz;
                const int dy = q % Ty;
                const int dx = q / Ty;
                const int gx = wrapi(bx0 + dx, X);
                const int gy = wrapi(by0 + dy, Y);
                const int gz = wrapi(bz0 + dz, Z);
                const int gidx = ((gx * Y + gy) * Z + gz) * 3 + c;
#if __has_builtin(__builtin_amdgcn_global_load_async_to_lds_b32)
                __builtin_amdgcn_global_load_async_to_lds_b32(
                    (GAS int*)(coeff + gidx), (LAS int*)(&tile[e]), 0, 0);
#else
                tile[e] = coeff[gidx];
#endif
            }
#if __has_builtin(__builtin_amdgcn_global_load_async_to_lds_b32)
#if __has_builtin(__builtin_amdgcn_s_wait_asynccnt)
            __builtin_amdgcn_s_wait_asynccnt(0);
#else
            asm volatile("s_wait_asynccnt 0" ::: "memory");
#endif
#endif
        }
        __syncthreads();

        const int i0 = o0 + vi0, i1 = o1 + vi1, i2 = o2 + vi2;
        if (i0 < fd0 && i1 < fd1 && i2 < fd2) {
            const float v0 = (float)i0, v1 = (float)i1, v2 = (float)i2;
            const float g0 = fa[0] * v0 + fa[1] * v1 + fa[2]  * v2 + fa[3];
            const float g1 = fa[4] * v0 + fa[5] * v1 + fa[6]  * v2 + fa[7];
            const float g2 = fa[8] * v0 + fa[9] * v1 + fa[10] * v2 + fa[11];
            const float f0 = floorf(g0), f1 = floorf(g1), f2 = floorf(g2);
            float wx[4], wy[4], wz[4];
            bw4(g0 - f0, wx); bw4(g1 - f1, wy); bw4(g2 - f2, wz);
            const int bx = (int)f0 - 1, by = (int)f1 - 1, bz = (int)f2 - 1;

            float acc0 = 0.0f, acc1 = 0.0f, acc2 = 0.0f;
            if (useLds) {
                const int lx = bx - bx0, ly = by - by0, lz = bz - bz0;
#pragma unroll
                for (int a = 0; a < 4; a++) {
                    float sy0 = 0.0f, sy1 = 0.0f, sy2 = 0.0f;
                    const int rowA = (lx + a) * Ty;
#pragma unroll
                    for (int b = 0; b < 4; b++) {
                        float sz0 = 0.0f, sz1 = 0.0f, sz2 = 0.0f;
                        const int rowB = (rowA + ly + b) * Tz + lz;
#pragma unroll
                        for (int c = 0; c < 4; c++) {
                            const float* p = &tile[(rowB + c) * 3];
                            sz0 = fmaf(wz[c], p[0], sz0);
                            sz1 = fmaf(wz[c], p[1], sz1);
                            sz2 = fmaf(wz[c], p[2], sz2);
                        }
                        sy0 = fmaf(wy[b], sz0, sy0);
                        sy1 = fmaf(wy[b], sz1, sy1);
                        sy2 = fmaf(wy[b], sz2, sy2);
                    }
                    acc0 = fmaf(wx[a], sy0, acc0);
                    acc1 = fmaf(wx[a], sy1, acc1);
                    acc2 = fmaf(wx[a], sy2, acc2);
                }
            } else {
                int ixw[4], iyw[4], izw[4];
#pragma unroll
                for (int a = 0; a < 4; a++) {
                    ixw[a] = wrapi(bx + a, X);
                    iyw[a] = wrapi(by + a, Y);
                    izw[a] = wrapi(bz + a, Z);
                }
#pragma unroll
                for (int a = 0; a < 4; a++) {
                    float sy0 = 0.0f, sy1 = 0.0f, sy2 = 0.0f;
#pragma unroll
                    for (int b = 0; b < 4; b++) {
                        float sz0 = 0.0f, sz1 = 0.0f, sz2 = 0.0f;
                        const int rowAB = (ixw[a] * Y + iyw[b]) * Z;
#pragma unroll
                        for (int c = 0; c < 4; c++) {
                            const float* p = &coeff[(rowAB + izw[c]) * 3];
                            sz0 = fmaf(wz[c], p[0], sz0);
                            sz1 = fmaf(wz[c], p[1], sz1);
                            sz2 = fmaf(wz[c], p[2], sz2);
                        }
                        sy0 = fmaf(wy[b], sz0, sy0);
                        sy1 = fmaf(wy[b], sz1, sy1);
                        sy2 = fmaf(wy[b], sz2, sy2);
                    }
                    acc0 = fmaf(wx[a], sy0, acc0);
                    acc1 = fmaf(wx[a], sy1, acc1);
                    acc2 = fmaf(wx[a], sy2, acc2);
                }
            }

            const float h0 = g0 + acc0, h1 = g1 + acc1, h2 = g2 + acc2;
            const float r0 = ma[0] * h0 + ma[1] * h1 + ma[2]  * h2 + ma[3];
            const float r1 = ma[4] * h0 + ma[5] * h1 + ma[6]  * h2 + ma[7];
            const float r2 = ma[8] * h0 + ma[9] * h1 + ma[10] * h2 + ma[11];

            const size_t oidx = (((size_t)i0 * fd1 + i1) * fd2 + i2) * 3;
            out[oidx + 0] = r0;
            out[oidx + 1] = r1;
            out[oidx + 2] = r2;
        }
    }
}

extern "C" void kernel_launch(void* const* d_in, const int* in_sizes, int n_in,
                              void* d_out, int out_size, void* d_ws, size_t ws_size,
                              hipStream_t stream) {
    (void)n_in; (void)ws_size;
    const float* coeff = (const float*)d_in[0];
    const float* A     = (const float*)d_in[1];
    const float* F     = (const float*)d_in[2];
    const float* M     = (const float*)d_in[3];
    const int*   pfd0  = (const int*)d_in[4];
    const int*   pfd1  = (const int*)d_in[5];
    const int*   pfd2  = (const int*)d_in[6];
    float* out  = (float*)d_out;
    float* mats = (float*)d_ws;

    // Recover cubic coeff grid dims from flat size (X*Y*Z*3, assumed cubic).
    const long long nvol = (long long)in_sizes[0] / 3;
    int X = (int)llround(cbrt((double)nvol));
    while ((long long)X * X * X < nvol) ++X;
    while (X > 1 && (long long)X * X * X > nvol) --X;

    setup_mats_kernel<<<1, 1, 0, stream>>>(A, F, M, mats);

    // fd0..2 live on device; persistent blocks loop over tiles in-kernel.
    long long blocks = (long long)(out_size / 3 + BLOCK - 1) / BLOCK + 16;
    if (blocks < 1) blocks = 1;
    if (blocks > (1 << 20)) blocks = (1 << 20);
    spline_disp_kernel<<<dim3((unsigned)blocks), dim3(BLOCK), 0, stream>>>(
        coeff, mats, pfd0, pfd1, pfd2, X, X, X, out);
}